// STPatchEmbedding_24421184045166
// MI455X (gfx1250) — compile-verified
//
#include <hip/hip_runtime.h>
#include <hip/hip_bf16.h>

#define B_DIM 2
#define N_DIM 207
#define NP_DIM 208              // N padded to 16
#define T_DIM 864
#define L_DIM 12
#define D_DIM 96
#define P_DIM 72
#define C_DIM (D_DIM * L_DIM)   // 1152
#define EPS_F 1e-8f
#define CT_PER_WAVE 4           // column tiles per wave in weff GEMM

typedef float v2f __attribute__((ext_vector_type(2)));
typedef float v8f __attribute__((ext_vector_type(8)));

// ---------------------------------------------------------------------------
// Kernel 1: Gumbel-softmax top-k mask -> zero-padded 208x208 buffer.
// Grid = 208 rows; row 207 and column 207 are written as zeros so the GEMM
// consumer needs no bounds checks.
// rank-based top-k: keep soft[j] iff (#{>}+#{== with smaller idx}) < k,
// matching jax.lax.top_k first-occurrence tie-breaking.
// ---------------------------------------------------------------------------
__global__ __launch_bounds__(256) void mask_topk_kernel(
    const float* __restrict__ adj, const float* __restrict__ noise,
    const int* __restrict__ kptr, float* __restrict__ maskP) {
  const int i = blockIdx.x;
  const int tid = threadIdx.x;
  if (i >= N_DIM) {                       // padding row
    if (tid < NP_DIM) maskP[i * NP_DIM + tid] = 0.f;
    return;
  }
  const int k = kptr[0];
  __shared__ float red[256];
  __shared__ float soft[N_DIM];

  float z = -3.4e38f;
  if (tid < N_DIM) {
    float a = adj[i * N_DIM + tid];
    a = a > 0.f ? a : 0.f;                // relu
    float logit = __logf(a + EPS_F);
    float u = noise[i * N_DIM + tid];
    float g = -__logf(-__logf(u + EPS_F)); // gumbel
    z = logit + g;                        // TAU = 1.0
  }
  red[tid] = z;
  __syncthreads();
  for (int s = 128; s > 0; s >>= 1) {
    if (tid < s) red[tid] = fmaxf(red[tid], red[tid + s]);
    __syncthreads();
  }
  const float zmax = red[0];
  __syncthreads();
  const float e = (tid < N_DIM) ? __expf(z - zmax) : 0.f;
  red[tid] = e;
  __syncthreads();
  for (int s = 128; s > 0; s >>= 1) {
    if (tid < s) red[tid] += red[tid + s];
    __syncthreads();
  }
  const float denom = red[0];
  __syncthreads();
  const float sv = e / denom;
  if (tid < N_DIM) soft[tid] = sv;
  __syncthreads();
  if (tid < NP_DIM) {
    float outv = 0.f;
    if (tid < N_DIM) {
      int rank = 0;
      for (int j = 0; j < N_DIM; ++j) {
        float o = soft[j];
        rank += (o > sv) || (o == sv && j < tid);
      }
      outv = (rank < k) ? sv : 0.f;
    }
    maskP[i * NP_DIM + tid] = outv;
  }
}

// ---------------------------------------------------------------------------
// Kernel 2: conv_weight (D,N,L) -> WrP (208, D*L), K-padding row zeroed.
// ---------------------------------------------------------------------------
__global__ __launch_bounds__(256) void transpose_w_kernel(
    const float* __restrict__ w, float* __restrict__ wrP) {
  int o = blockIdx.x * 256 + threadIdx.x;
  if (o >= NP_DIM * C_DIM) return;
  int j = o / C_DIM;
  int c = o % C_DIM;
  int d = c / L_DIM, l = c % L_DIM;
  wrP[o] = (j < N_DIM) ? w[(d * N_DIM + j) * L_DIM + l] : 0.f;
}

// ---------------------------------------------------------------------------
// Kernel 3: WeffP = maskP @ WrP  (208x208 @ 208x1152), V_WMMA_F32_16X16X4_F32.
// One wave per (16-row x 64-col) strip: 4 column tiles share each A fragment.
// All operands zero-padded -> branch-free inner loop (pure loads + wmma).
// ---------------------------------------------------------------------------
__global__ __launch_bounds__(32) void weff_gemm_kernel(
    const float* __restrict__ maskP, const float* __restrict__ wrP,
    float* __restrict__ weffP) {
  const int ctBase = blockIdx.x * CT_PER_WAVE; // column tile group
  const int mt = blockIdx.y;                   // row tile: 0..12
  const int lane = threadIdx.x;
  const int m = lane & 15;
  const int half = lane >> 4;
  const int row = mt * 16 + m;                 // 0..207, in-bounds of maskP

  const float* __restrict__ arow = maskP + row * NP_DIM;

  v8f acc0 = {}, acc1 = {}, acc2 = {}, acc3 = {};
  const int col0 = ctBase * 16 + m;

  for (int k0 = 0; k0 < NP_DIM; k0 += 4) {
    const int ka = k0 + 2 * half;
    v2f a;
    a.x = arow[ka];
    a.y = arow[ka + 1];
    const float* __restrict__ b0 = wrP + ka * C_DIM + col0;
    const float* __restrict__ b1 = wrP + (ka + 1) * C_DIM + col0;
    v2f b;
    b.x = b0[0];       b.y = b1[0];
    acc0 = __builtin_amdgcn_wmma_f32_16x16x4_f32(false, a, false, b, (short)0, acc0, false, false);
    b.x = b0[16];      b.y = b1[16];
    acc1 = __builtin_amdgcn_wmma_f32_16x16x4_f32(false, a, false, b, (short)0, acc1, false, false);
    b.x = b0[32];      b.y = b1[32];
    acc2 = __builtin_amdgcn_wmma_f32_16x16x4_f32(false, a, false, b, (short)0, acc2, false, false);
    b.x = b0[48];      b.y = b1[48];
    acc3 = __builtin_amdgcn_wmma_f32_16x16x4_f32(false, a, false, b, (short)0, acc3, false, false);
  }

  // C/D layout: VGPR r -> M = r + 8*half, N = lane&15. weffP is padded: no guards.
  for (int r = 0; r < 8; ++r) {
    const int orow = mt * 16 + r + 8 * half;
    float* __restrict__ orow_p = weffP + orow * C_DIM + col0;
    orow_p[0]  = acc0[r];
    orow_p[16] = acc1[r];
    orow_p[32] = acc2[r];
    orow_p[48] = acc3[r];
  }
}

// ---------------------------------------------------------------------------
// Kernel 4: per-(b,i) patch GEMM: out[b,i,d,p] = sum_l x[p*12+l]*Weff[i,d,l]+bias.
// M = P = 72 (padded to 80 via zeroed LDS tail), N = D = 96 (6 waves, one
// column tile each), K = 12 = 3 steps of 4. WMMA in uniform control flow.
// ---------------------------------------------------------------------------
__global__ __launch_bounds__(192) void patch_gemm_kernel(
    const float* __restrict__ x, const float* __restrict__ weffP,
    const float* __restrict__ bias, float* __restrict__ out) {
  const int bi = blockIdx.x;            // 0 .. B*N-1
  const int i = bi % N_DIM;
  const int tid = threadIdx.x;
  const int wave = tid >> 5;            // 0..5 -> d tile
  const int lane = tid & 31;
  const int m = lane & 15;
  const int half = lane >> 4;

  __shared__ float xl[960];             // 80 padded rows * 12
  __shared__ float wl[C_DIM];
  __shared__ float bl[D_DIM];

  for (int idx = tid; idx < 960; idx += 192)
    xl[idx] = (idx < T_DIM) ? x[(size_t)bi * T_DIM + idx] : 0.f;
  for (int idx = tid; idx < C_DIM; idx += 192)
    wl[idx] = weffP[(size_t)i * C_DIM + idx];
  if (tid < D_DIM) bl[tid] = bias[tid];
  __syncthreads();

  const int d = wave * 16 + m;
  const float bv = bl[d];
  for (int mt = 0; mt < 5; ++mt) {
    v8f acc = {};
    const int prow = mt * 16 + m;
#pragma unroll
    for (int ks = 0; ks < 3; ++ks) {
      const int ka = ks * 4 + 2 * half;
      v2f a, b;
      a.x = xl[prow * L_DIM + ka];
      a.y = xl[prow * L_DIM + ka + 1];
      b.x = wl[d * L_DIM + ka];
      b.y = wl[d * L_DIM + ka + 1];
      acc = __builtin_amdgcn_wmma_f32_16x16x4_f32(
          false, a, false, b, (short)0, acc, false, false);
    }
#pragma unroll
    for (int r = 0; r < 8; ++r) {
      const int p = mt * 16 + r + 8 * half;
      if (p < P_DIM) out[((size_t)bi * D_DIM + d) * P_DIM + p] = acc[r] + bv;
    }
  }
}

// ---------------------------------------------------------------------------
extern "C" void kernel_launch(void* const* d_in, const int* in_sizes, int n_in,
                              void* d_out, int out_size, void* d_ws, size_t ws_size,
                              hipStream_t stream) {
  const float* x     = (const float*)d_in[0];  // (B,N,1,T)
  const float* adj   = (const float*)d_in[1];  // (N,N)
  const float* noise = (const float*)d_in[2];  // (N,N)
  const float* w     = (const float*)d_in[3];  // (D,N,L)
  const float* bias  = (const float*)d_in[4];  // (D,)
  const int*   kptr  = (const int*)d_in[5];    // scalar k

  float* maskP = (float*)d_ws;                       // 208*208
  float* wrP   = maskP + NP_DIM * NP_DIM;            // 208*1152
  float* weffP = wrP + NP_DIM * C_DIM;               // 208*1152
  float* out   = (float*)d_out;

  mask_topk_kernel<<<NP_DIM, 256, 0, stream>>>(adj, noise, kptr, maskP);
  transpose_w_kernel<<<(NP_DIM * C_DIM + 255) / 256, 256, 0, stream>>>(w, wrP);
  weff_gemm_kernel<<<dim3(C_DIM / (16 * CT_PER_WAVE), NP_DIM / 16), 32, 0, stream>>>(
      maskP, wrP, weffP);
  patch_gemm_kernel<<<B_DIM * N_DIM, 192, 0, stream>>>(x, weffP, bias, out);
}